// ROUGELoss_3822520893773
// MI455X (gfx1250) — compile-verified
//
#include <hip/hip_runtime.h>

#define MAX_LEN 128
#define PAD_ID  1
#define VSZ     32000
#define BSZ     16

typedef __attribute__((ext_vector_type(2))) float v2f;
typedef __attribute__((ext_vector_type(4))) float v4f;
typedef __attribute__((ext_vector_type(8))) float v8f;

// exp((logit + g)/tau) with tau=1, g = -log(-log u), u clipped like the reference:
// = exp(logit) * exp(-log(-log u)) = exp(logit) / (-log u)
__device__ __forceinline__ float gumbel_exp(float logit, float u) {
    u = fminf(fmaxf(u, 1e-10f), 1.0f - 1e-7f);
    return __expf(logit) / (-__logf(u));
}

// ---------------------------------------------------------------------------
// Kernel 1: one block per (b, j) row. Streaming softmax-denominator reduction
// over V=32000 (memory bound: 524 MB total -> ~22.5us at 23.3 TB/s), then
// gather the 128 label positions and scatter O[b][i][j] = m_i*m_j*prob.
// ---------------------------------------------------------------------------
__global__ __launch_bounds__(256) void rouge_rows(
        const float* __restrict__ logits,
        const int*   __restrict__ labels,
        const float* __restrict__ gumbel,
        float*       __restrict__ O) {
    const int row = blockIdx.x;              // 0..2047
    const int b   = row >> 7;
    const int j   = row & (MAX_LEN - 1);
    const int tid = threadIdx.x;

    const v4f* lg4 = reinterpret_cast<const v4f*>(logits + (size_t)row * VSZ);
    const v4f* gu4 = reinterpret_cast<const v4f*>(gumbel + (size_t)row * VSZ);

    float part = 0.0f;
    for (int i = tid; i < VSZ / 4; i += 256) {
        v4f l = __builtin_nontemporal_load(&lg4[i]);   // single-use stream: NT hint
        v4f u = __builtin_nontemporal_load(&gu4[i]);
        part += gumbel_exp(l.x, u.x) + gumbel_exp(l.y, u.y)
              + gumbel_exp(l.z, u.z) + gumbel_exp(l.w, u.w);
    }

    // wave32 reduce, then cross-wave via LDS
    #pragma unroll
    for (int off = 16; off; off >>= 1) part += __shfl_down(part, off, 32);

    __shared__ float wsum[8];
    __shared__ float stot;
    if ((tid & 31) == 0) wsum[tid >> 5] = part;
    __syncthreads();
    if (tid == 0) {
        float s = 0.0f;
        #pragma unroll
        for (int w = 0; w < 8; ++w) s += wsum[w];
        stot = s;
    }
    __syncthreads();

    const float inv_s = 1.0f / stot;
    const float mj = (labels[b * MAX_LEN + j] != PAD_ID) ? 1.0f : 0.0f;

    if (tid < MAX_LEN) {
        const int   li  = labels[b * MAX_LEN + tid];
        const float mi  = (li != PAD_ID) ? 1.0f : 0.0f;
        const float lgv = logits[(size_t)row * VSZ + li];
        const float uv  = gumbel[(size_t)row * VSZ + li];
        const float val = mi * mj * gumbel_exp(lgv, uv) * inv_s;
        // O[b][i][j], i = tid (column-scatter; tiny vs. main traffic)
        O[((size_t)b * MAX_LEN + tid) * MAX_LEN + j] = val;
    }
}

// ---------------------------------------------------------------------------
// Kernel 2: one wave per batch. sudoku + sum collapses to:
//   r_i = max(rowsum_i, 1);  colsum_j = sum_i O[i][j]/r_i
//   total = sum_j colsum_j / max(colsum_j, 1)
// Column sums computed with V_WMMA_F32_16X16X4_F32: D = ones(16x4)*B(4x16)+C.
// A==ones makes the K-row assignment permutation-invariant; N = lane&15.
// ---------------------------------------------------------------------------
__global__ __launch_bounds__(32) void rouge_sudoku(
        const float* __restrict__ O,
        const int*   __restrict__ labels,
        float*       __restrict__ partial) {
    const int b    = blockIdx.x;
    const int lane = threadIdx.x;
    const float* Ob = O + (size_t)b * MAX_LEN * MAX_LEN;

    __shared__ float inv_r[MAX_LEN];

    // row sums: 4 rows per lane, vectorized
    #pragma unroll
    for (int rr = 0; rr < 4; ++rr) {
        const int r = lane * 4 + rr;
        const v4f* row4 = reinterpret_cast<const v4f*>(Ob + (size_t)r * MAX_LEN);
        float s = 0.0f;
        #pragma unroll 8
        for (int c = 0; c < MAX_LEN / 4; ++c) {
            v4f v = row4[c];
            s += v.x + v.y + v.z + v.w;
        }
        inv_r[r] = 1.0f / fmaxf(s, 1.0f);
    }
    __syncthreads();

    v2f ones; ones.x = 1.0f; ones.y = 1.0f;   // A = 16x4 all-ones (2 VGPRs/lane)
    const int n     = lane & 15;               // output column within 16-wide group
    const int khalf = lane >> 4;               // which pair of K rows this lane feeds

    float total = 0.0f;
    for (int jg = 0; jg < MAX_LEN / 16; ++jg) {
        v8f acc = {};
        const int jcol = jg * 16 + n;
        for (int i0 = 0; i0 < MAX_LEN; i0 += 4) {
            const int r0 = i0 + khalf * 2;
            v2f bv;
            bv.x = Ob[(size_t)r0       * MAX_LEN + jcol] * inv_r[r0];
            bv.y = Ob[(size_t)(r0 + 1) * MAX_LEN + jcol] * inv_r[r0 + 1];
            acc = __builtin_amdgcn_wmma_f32_16x16x4_f32(
                    /*neg_a=*/false, ones, /*neg_b=*/false, bv,
                    /*c_mod=*/(short)0, acc, /*reuse_a=*/false, /*reuse_b=*/false);
        }
        // every D row holds colsum; lanes 16-31 duplicate lanes 0-15 -> mask them
        const float cs = acc[0];
        total += (lane < 16) ? (cs / fmaxf(cs, 1.0f)) : 0.0f;
    }

    #pragma unroll
    for (int off = 16; off; off >>= 1) total += __shfl_down(total, off, 32);

    if (lane == 0) {
        int cnt = 0;
        for (int i = 0; i < MAX_LEN; ++i) cnt += (labels[b * MAX_LEN + i] != PAD_ID);
        partial[b] = -total / (float)cnt;
    }
}

// ---------------------------------------------------------------------------
// Kernel 3: average the 16 per-batch results.
// ---------------------------------------------------------------------------
__global__ __launch_bounds__(32) void rouge_final(
        const float* __restrict__ partial, float* __restrict__ out) {
    float v = (threadIdx.x < BSZ) ? partial[threadIdx.x] : 0.0f;
    #pragma unroll
    for (int off = 16; off; off >>= 1) v += __shfl_down(v, off, 32);
    if (threadIdx.x == 0) out[0] = v / (float)BSZ;
}

extern "C" void kernel_launch(void* const* d_in, const int* in_sizes, int n_in,
                              void* d_out, int out_size, void* d_ws, size_t ws_size,
                              hipStream_t stream) {
    const float* logits = (const float*)d_in[0];   // (16,128,32000) f32
    const int*   labels = (const int*)d_in[1];     // (16,128) i32
    const float* gumbel = (const float*)d_in[2];   // (16,128,32000) f32

    float* O       = (float*)d_ws;                          // 16*128*128 f32 = 1 MB
    float* partial = O + (size_t)BSZ * MAX_LEN * MAX_LEN;   // 16 f32

    rouge_rows  <<<BSZ * MAX_LEN, 256, 0, stream>>>(logits, labels, gumbel, O);
    rouge_sudoku<<<BSZ,           32,  0, stream>>>(O, labels, partial);
    rouge_final <<<1,             32,  0, stream>>>(partial, (float*)d_out);
}